// LogEigAJD_24129126269562
// MI455X (gfx1250) — compile-verified
//
#include <hip/hip_runtime.h>

// ---------------------------------------------------------------------------
// Batched matrix-log of SPD matrices on gfx1250 (MI455X).
//   out = V * diag(log(relu(w)+eps)) * V^T   per 64x64 matrix.
// One workgroup (128 threads / 4 wave32) per matrix:
//   1) LDS-resident cyclic Jacobi eigensolver (f32 VALU)
//   2) reconstruction GEMM via V_WMMA_F32_16X16X4_F32 (the only f32 WMMA)
// ---------------------------------------------------------------------------

typedef __attribute__((ext_vector_type(2))) float v2f;
typedef __attribute__((ext_vector_type(8))) float v8f;

#define NDIM     64
#define LDA      66          // padded LDS row stride (floats) -> no bank conflicts on column walks
#define NTHREADS 128         // 4 waves of 32
#define NSWEEPS  7           // cyclic Jacobi sweeps (63 rounds each)
#define LOG_EPS  1e-9f

__global__ __launch_bounds__(NTHREADS)
void logeig_jacobi_wmma(const float* __restrict__ in,
                        float* __restrict__ out)
{
    __shared__ float A[NDIM * LDA];   // working matrix, later W = V*diag(log s)
    __shared__ float V[NDIM * LDA];   // accumulated eigenvectors (columns)
    __shared__ float cs[32], sn[32];  // per-pair rotation
    __shared__ int   pp[32], qq[32];  // normalized pair indices (p<q)
    __shared__ int   top[32], bot[32];// round-robin tournament state
    __shared__ float ls[NDIM];        // log-eigenvalues

    const int tid = threadIdx.x;
    const float* __restrict__ src = in  + (size_t)blockIdx.x * (NDIM * NDIM);
    float*       __restrict__ dst = out + (size_t)blockIdx.x * (NDIM * NDIM);

    // ---- load A (float4 vectorized -> global_load_b128), init V = I --------
    for (int i = tid; i < (NDIM * NDIM) / 4; i += NTHREADS) {
        float4 vv = reinterpret_cast<const float4*>(src)[i];
        const int r = i >> 4;            // (i*4)/64
        const int c = (i & 15) << 2;     // (i*4)%64
        float* Ar = &A[r * LDA + c];
        Ar[0] = vv.x; Ar[1] = vv.y; Ar[2] = vv.z; Ar[3] = vv.w;
        float* Vr = &V[r * LDA + c];
        Vr[0] = (r == c + 0) ? 1.0f : 0.0f;
        Vr[1] = (r == c + 1) ? 1.0f : 0.0f;
        Vr[2] = (r == c + 2) ? 1.0f : 0.0f;
        Vr[3] = (r == c + 3) ? 1.0f : 0.0f;
    }
    if (tid < 32) { top[tid] = 2 * tid; bot[tid] = 2 * tid + 1; }
    __syncthreads();

    // ---- cyclic Jacobi: 32 disjoint pairs per round, 63 rounds per sweep ---
    for (int sweep = 0; sweep < NSWEEPS; ++sweep) {
        for (int round = 0; round < 2 * NDIM / 2 - 1; ++round) {  // 63 rounds
            // (a) 32 lanes compute the Givens rotations
            if (tid < 32) {
                int p = top[tid], q = bot[tid];
                if (p > q) { int t = p; p = q; q = t; }
                pp[tid] = p; qq[tid] = q;
                const float app = A[p * LDA + p];
                const float aqq = A[q * LDA + q];
                const float apq = A[p * LDA + q];
                float c = 1.0f, s = 0.0f;
                if (__builtin_fabsf(apq) > 1e-12f) {
                    const float tau = (aqq - app) / (2.0f * apq);
                    const float t   = __builtin_copysignf(1.0f, tau) /
                                      (__builtin_fabsf(tau) + sqrtf(1.0f + tau * tau));
                    c = 1.0f / sqrtf(1.0f + t * t);
                    s = t * c;
                }
                cs[tid] = c; sn[tid] = s;
            }
            __syncthreads();

            // (b) left update: A <- J^T A   (rows p,q of all 32 pairs)
            for (int w = tid; w < 32 * NDIM; w += NTHREADS) {
                const int pair = w >> 6, k = w & 63;
                const int p = pp[pair], q = qq[pair];
                const float c = cs[pair], s = sn[pair];
                const float apk = A[p * LDA + k], aqk = A[q * LDA + k];
                A[p * LDA + k] = c * apk - s * aqk;
                A[q * LDA + k] = s * apk + c * aqk;
            }
            __syncthreads();

            // (c) right update: A <- A J, eigvec accumulate: V <- V J
            for (int w = tid; w < 32 * NDIM; w += NTHREADS) {
                const int pair = w >> 6, k = w & 63;
                const int p = pp[pair], q = qq[pair];
                const float c = cs[pair], s = sn[pair];
                const float akp = A[k * LDA + p], akq = A[k * LDA + q];
                A[k * LDA + p] = c * akp - s * akq;
                A[k * LDA + q] = s * akp + c * akq;
                const float vkp = V[k * LDA + p], vkq = V[k * LDA + q];
                V[k * LDA + p] = c * vkp - s * vkq;
                V[k * LDA + q] = s * vkp + c * vkq;
            }

            // (d) round-robin rotate (Brent–Luk tournament), element 0 fixed
            int t = 0, b = 0;
            if (tid < 32) { t = top[tid]; b = bot[tid]; }
            __syncthreads();   // also fences (c)'s LDS writes
            if (tid >= 1 && tid <= 30) top[tid + 1] = t;
            if (tid == 0)              top[1]       = b;
            if (tid >= 1 && tid <= 31) bot[tid - 1] = b;
            if (tid == 31)             bot[31]      = t;
            __syncthreads();
        }
    }

    // ---- spectral log, then W = V * diag(log s) built in place of A --------
    if (tid < NDIM) {
        const float lam = A[tid * LDA + tid];
        ls[tid] = logf(fmaxf(lam, 0.0f) + LOG_EPS);
    }
    __syncthreads();
    for (int i = tid; i < NDIM * NDIM; i += NTHREADS) {
        const int r = i >> 6, c = i & 63;
        A[r * LDA + c] = V[r * LDA + c] * ls[c];
    }
    __syncthreads();

    // ---- reconstruction: out = W @ V^T via V_WMMA_F32_16X16X4_F32 ----------
    // Fragment layouts per CDNA5 ISA 7.12.2 (32-bit operands):
    //   A 16x4 : lane L (M = L%16), group g = L/16 -> vgpr v holds K = 2g+v
    //   B 4x16 : lane L (N = L%16), group g        -> vgpr v holds K = 2g+v
    //   C 16x16: vgpr r holds M = r + 8g, N = L%16
    // wave handles row-block `wave`, loops over the four 16-col blocks.
    const int wave = tid >> 5;
    const int lane = tid & 31;
    const int g    = lane >> 4;
    const int l    = lane & 15;
    const int rowA = (wave * 16 + l) * LDA;   // W rows for this wave's tile

    for (int u = 0; u < 4; ++u) {
        const int rowB = (u * 16 + l) * LDA;  // V rows: B[k][n] = V[n][k]
        v8f acc = {};
#pragma unroll
        for (int kb = 0; kb < NDIM; kb += 4) {
            const int k0 = kb + (g << 1);
            v2f a = { A[rowA + k0], A[rowA + k0 + 1] };
            v2f b = { V[rowB + k0], V[rowB + k0 + 1] };
            acc = __builtin_amdgcn_wmma_f32_16x16x4_f32(
                      /*neg_a=*/false, a, /*neg_b=*/false, b,
                      /*c_mod=*/(short)0, acc,
                      /*reuse_a=*/false, /*reuse_b=*/false);
        }
#pragma unroll
        for (int r = 0; r < 8; ++r)
            dst[(wave * 16 + r + (g << 3)) * NDIM + u * 16 + l] = acc[r];
    }
}

extern "C" void kernel_launch(void* const* d_in, const int* in_sizes, int n_in,
                              void* d_out, int out_size, void* d_ws, size_t ws_size,
                              hipStream_t stream) {
    (void)n_in; (void)out_size; (void)d_ws; (void)ws_size;
    const float* in = (const float*)d_in[0];
    float* out = (float*)d_out;
    const int nmat = in_sizes[0] / (NDIM * NDIM);   // 256*32 = 8192
    logeig_jacobi_wmma<<<dim3(nmat), dim3(NTHREADS), 0, stream>>>(in, out);
}